// SimpleBlockMultiGraph_4054449127564
// MI455X (gfx1250) — compile-verified
//
#include <hip/hip_runtime.h>
#include <hip/hip_bf16.h>

typedef float v2f __attribute__((ext_vector_type(2)));
typedef float v8f __attribute__((ext_vector_type(8)));

#define N_NODES 100000
#define N_EDGES 1600000
#define WSTEPS  12
#define CIN     16
#define CMID    32
#define COUT    32
#define OUT_STRIDE (WSTEPS * COUT)   // 384 floats per node in d_out
#define LDS_NODE_STRIDE 388          // padded to avoid 16-way LDS bank conflicts

// ---------------------------------------------------------------- zero fill
__global__ void zero_f4_kernel(float4* __restrict__ p, long long n4) {
    long long i = (long long)blockIdx.x * blockDim.x + threadIdx.x;
    if (i < n4) p[i] = make_float4(0.f, 0.f, 0.f, 0.f);
}

// ---------------------------------------------------------------- degree
__global__ void degree_kernel(const int* __restrict__ col,
                              const float* __restrict__ ew,
                              float* __restrict__ deg) {
    int e = blockIdx.x * blockDim.x + threadIdx.x;
    if (e < N_EDGES) atomicAdd(&deg[col[e]], ew[e]);
}

// ---------------------------------------------------------------- edge norm
__device__ __forceinline__ float inv_sqrt_guard(float d) {
    return d > 0.f ? rsqrtf(d) : 0.f;
}
__global__ void norm_kernel(const int* __restrict__ row,
                            const int* __restrict__ col,
                            const float* __restrict__ ew,
                            const float* __restrict__ deg,
                            float* __restrict__ norm) {
    int e = blockIdx.x * blockDim.x + threadIdx.x;
    if (e < N_EDGES) {
        float dr = deg[row[e]];
        float dc = deg[col[e]];
        norm[e] = inv_sqrt_guard(dr) * ew[e] * inv_sqrt_guard(dc);
    }
}

// ---------------------------------------------------------------- GCN GEMM: h = x_t @ W_t  (WMMA f32 16x16x4)
// x_t: [N,16] row-major, w_t: [16,32] row-major, h: [N,32] row-major.
__global__ void gcn_gemm_wmma(const float* __restrict__ x_t,
                              const float* __restrict__ w_t,
                              float* __restrict__ h) {
    int wave = (blockIdx.x * blockDim.x + threadIdx.x) >> 5;
    int lane = threadIdx.x & 31;
    int base = wave * 16;                 // 16 nodes per wave
    if (base >= N_NODES) return;          // uniform per wave -> EXEC stays all-1s
    int m    = lane & 15;                 // row within tile
    int koff = (lane >> 4) * 2;           // K sub-offset: 0 (lanes 0-15) / 2 (lanes 16-31)

    v8f acc0 = {};                        // cols 0..15
    v8f acc1 = {};                        // cols 16..31
    const float* xrow = x_t + (size_t)(base + m) * CIN;

#pragma unroll
    for (int k0 = 0; k0 < CIN; k0 += 4) {
        v2f a = *(const v2f*)(xrow + k0 + koff);      // K=k0+koff, k0+koff+1
        v2f b0, b1;
        b0.x = w_t[(k0 + koff)     * CMID + m];
        b0.y = w_t[(k0 + koff + 1) * CMID + m];
        b1.x = w_t[(k0 + koff)     * CMID + 16 + m];
        b1.y = w_t[(k0 + koff + 1) * CMID + 16 + m];
        acc0 = __builtin_amdgcn_wmma_f32_16x16x4_f32(false, a, false, b0, (short)0, acc0, false, false);
        acc1 = __builtin_amdgcn_wmma_f32_16x16x4_f32(false, a, false, b1, (short)0, acc1, false, false);
    }
    int mrow = (lane >> 4) * 8;           // D rows: j (+8 for upper half-wave)
#pragma unroll
    for (int j = 0; j < 8; ++j) {
        size_t r = (size_t)(base + mrow + j) * CMID;
        h[r + m]      = acc0[j];
        h[r + 16 + m] = acc1[j];
    }
}

// ---------------------------------------------------------------- scatter: out[col, t, :] += h[row, :] * norm[e]
__global__ void gcn_scatter_kernel(const float* __restrict__ h,
                                   const int* __restrict__ row,
                                   const int* __restrict__ col,
                                   const float* __restrict__ norm,
                                   float* __restrict__ out, int t) {
    long long idx = (long long)blockIdx.x * blockDim.x + threadIdx.x;
    if (idx >= (long long)N_EDGES * 4) return;
    int e = (int)(idx >> 2);
    int q = (int)(idx & 3);               // 8-channel group
    float nv = norm[e];
    int r = row[e], c = col[e];
    const float4* hp = (const float4*)(h + (size_t)r * CMID + q * 8);
    float4 h0 = hp[0], h1 = hp[1];
    float* op = out + (size_t)c * OUT_STRIDE + t * CMID + q * 8;
    atomicAdd(op + 0, h0.x * nv);
    atomicAdd(op + 1, h0.y * nv);
    atomicAdd(op + 2, h0.z * nv);
    atomicAdd(op + 3, h0.w * nv);
    atomicAdd(op + 4, h1.x * nv);
    atomicAdd(op + 5, h1.y * nv);
    atomicAdd(op + 6, h1.z * nv);
    atomicAdd(op + 7, h1.w * nv);
}

// ---------------------------------------------------------------- temporal conv as 12 WMMA GEMMs per 16-node tile
// In-place on d_out: stage agg(+gcn_b) for 32 nodes in LDS, then
// out[:,w,:] = LeakyReLU( sum_k agg[:, w+k-1, :] @ Bk + conv_b ),  Bk[ci][co] = conv_w[co,ci,k]
__global__ __launch_bounds__(64)
void temporal_conv_wmma(float* __restrict__ out,
                        const float* __restrict__ gcnb,   // [12,32] flat 384
                        const float* __restrict__ cw,     // [32,32,3]
                        const float* __restrict__ cb) {   // [32]
    __shared__ float lds[32 * LDS_NODE_STRIDE];
    const int tid      = threadIdx.x;        // 0..63  (2 waves)
    const int lane     = tid & 31;
    const int wv       = tid >> 5;           // wave in block: 0/1
    const int nodeBase = blockIdx.x * 32;    // 32 nodes per block (N divisible)
    const int nlane    = lane & 15;          // column (co) / row (m) index in tile
    const int koff     = (lane >> 4) * 2;    // K sub-offset 0/2
    const int mrow     = (lane >> 4) * 8;    // D row base

    // ---- stage agg + gcn bias into LDS (float4, padded node stride) ----
    for (int nl = 0; nl < 32; ++nl) {
        const float4* src = (const float4*)(out + (size_t)(nodeBase + nl) * OUT_STRIDE);
        const float4* gb4 = (const float4*)gcnb;
        for (int r4 = tid; r4 < OUT_STRIDE / 4; r4 += 64) {
            float4 v = src[r4];
            float4 g = gb4[r4];
            v.x += g.x; v.y += g.y; v.z += g.z; v.w += g.w;
            *(float4*)&lds[nl * LDS_NODE_STRIDE + r4 * 4] = v;
        }
    }
    __syncthreads();

    // ---- hoist conv-weight B fragments to registers: [tap][k0][ntile] ----
    v2f Bf[3][8][2];
#pragma unroll
    for (int k = 0; k < 3; ++k)
#pragma unroll
        for (int k0 = 0; k0 < 8; ++k0)
#pragma unroll
            for (int nt = 0; nt < 2; ++nt) {
                int ci0 = k0 * 4 + koff;
                int co  = nt * 16 + nlane;
                Bf[k][k0][nt].x = cw[co * (CMID * 3) + ci0       * 3 + k];
                Bf[k][k0][nt].y = cw[co * (CMID * 3) + (ci0 + 1) * 3 + k];
            }
    const float bias0 = cb[nlane];
    const float bias1 = cb[16 + nlane];
    const float* arow = &lds[(wv * 16 + nlane) * LDS_NODE_STRIDE];   // node m = nlane of this wave

    for (int w = 0; w < WSTEPS; ++w) {
        v8f acc0, acc1;
#pragma unroll
        for (int j = 0; j < 8; ++j) { acc0[j] = bias0; acc1[j] = bias1; }

#pragma unroll
        for (int k = 0; k < 3; ++k) {
            int wt = w + k - 1;
            if (wt < 0 || wt >= WSTEPS) continue;      // uniform branch, EXEC stays full
            const float* src = arow + wt * CMID;
#pragma unroll
            for (int k0 = 0; k0 < 8; ++k0) {
                v2f a = *(const v2f*)(src + k0 * 4 + koff);   // ds_load_b64, 8B aligned
                acc0 = __builtin_amdgcn_wmma_f32_16x16x4_f32(false, a, false, Bf[k][k0][0], (short)0, acc0, false, false);
                acc1 = __builtin_amdgcn_wmma_f32_16x16x4_f32(false, a, false, Bf[k][k0][1], (short)0, acc1, false, false);
            }
        }
        // ---- LeakyReLU + in-place store (LDS already holds all inputs) ----
#pragma unroll
        for (int j = 0; j < 8; ++j) {
            int n = nodeBase + wv * 16 + mrow + j;
            float v0 = acc0[j], v1 = acc1[j];
            v0 = v0 >= 0.f ? v0 : 0.01f * v0;
            v1 = v1 >= 0.f ? v1 : 0.01f * v1;
            size_t o = (size_t)n * OUT_STRIDE + w * COUT;
            out[o + nlane]      = v0;
            out[o + 16 + nlane] = v1;
        }
    }
}

// ---------------------------------------------------------------- launch
extern "C" void kernel_launch(void* const* d_in, const int* in_sizes, int n_in,
                              void* d_out, int out_size, void* d_ws, size_t ws_size,
                              hipStream_t stream) {
    const float* x     = (const float*)d_in[0];   // [12, N, 16]
    const int*   A     = (const int*)  d_in[1];   // [2, E]
    const float* ew    = (const float*)d_in[2];   // [E]
    const float* gcn_w = (const float*)d_in[3];   // [12, 16, 32]
    const float* gcn_b = (const float*)d_in[4];   // [12, 32]
    const float* cw    = (const float*)d_in[5];   // [32, 32, 3]
    const float* cb    = (const float*)d_in[6];   // [32]
    float* out = (float*)d_out;                   // [N, 12, 32]

    const int* row = A;
    const int* col = A + N_EDGES;

    // workspace layout (floats): deg | norm | h
    float* wsf  = (float*)d_ws;
    float* deg  = wsf;                       // N floats
    float* norm = wsf + 131072;              // E floats
    float* h    = wsf + 131072 + 1638400;    // N*32 floats

    // zero d_out (scatter target) and deg
    {
        long long n4 = (long long)N_NODES * OUT_STRIDE / 4;   // 9.6M float4
        int blocks = (int)((n4 + 255) / 256);
        zero_f4_kernel<<<blocks, 256, 0, stream>>>((float4*)out, n4);
        long long d4 = N_NODES / 4;
        zero_f4_kernel<<<(int)((d4 + 255) / 256), 256, 0, stream>>>((float4*)deg, d4);
    }
    // degree + edge norm
    degree_kernel<<<(N_EDGES + 255) / 256, 256, 0, stream>>>(col, ew, deg);
    norm_kernel<<<(N_EDGES + 255) / 256, 256, 0, stream>>>(row, col, ew, deg, norm);

    // per-timestep GCN: WMMA GEMM then normalized scatter-add
    const int gemm_waves  = N_NODES / 16;                    // 6250
    const int gemm_blocks = (gemm_waves + 7) / 8;            // 256 threads = 8 waves
    const long long scat_threads = (long long)N_EDGES * 4;
    const int scat_blocks = (int)((scat_threads + 255) / 256);
    for (int t = 0; t < WSTEPS; ++t) {
        const float* x_t = x + (size_t)t * N_NODES * CIN;
        const float* w_t = gcn_w + (size_t)t * CIN * CMID;
        gcn_gemm_wmma<<<gemm_blocks, 256, 0, stream>>>(x_t, w_t, h);
        gcn_scatter_kernel<<<scat_blocks, 256, 0, stream>>>(h, row, col, norm, out, t);
    }

    // temporal conv + conv bias + LeakyReLU, in place on d_out
    temporal_conv_wmma<<<N_NODES / 32, 64, 0, stream>>>(out, gcn_b, cw, cb);
}